// ContrastiveLoss_4157528343143
// MI455X (gfx1250) — compile-verified
//
#include <hip/hip_runtime.h>
#include <hip/hip_bf16.h>
#include <math.h>

// CDNA5 WMMA fragment types
typedef float v2f __attribute__((ext_vector_type(2)));   // f32 16x4 A / 4x16 B frag (2 VGPRs)
typedef float v8f __attribute__((ext_vector_type(8)));   // f32 16x16 C/D frag (8 VGPRs)

#define D_DIM          128
#define TILE_PAIRS     16
#define ROWS_PER_TILE  32          // 16 pairs x 2 rows (interleaved i,j in the index array)
#define LDS_STRIDE     132         // dwords per staged row: 128 + 4 pad (16B-aligned, de-conflicts banks)
#define WAVES_PER_BLOCK 2
#define BLOCK_THREADS  (WAVES_PER_BLOCK * 32)

// One kernel handles either loss term (mode 0 = positive, mode 1 = negative).
// Per 16-pair tile:
//   stage 32 rows (512B each) into LDS via b128 loads, one row per lane;
//   run 32 chained V_WMMA_F32_16X16X4_F32 ops with A = (xi-xj)^2 frags and
//   B = ones, so acc[r] ends up holding dist^2 of pair r (lanes 0-15) or
//   pair r+8 (lanes 16-31), replicated across the 16 lanes of each half.
__global__ __launch_bounds__(BLOCK_THREADS)
void contrastive_wmma_kernel(const float* __restrict__ Xemb,
                             const int*   __restrict__ pairIdx,   // [P][2] flat
                             const float* __restrict__ h_bias,
                             float*       __restrict__ out,
                             int numPairs, int mode)
{
    __shared__ float smem[WAVES_PER_BLOCK][ROWS_PER_TILE][LDS_STRIDE];

    const int lane    = threadIdx.x & 31;
    const int warp    = threadIdx.x >> 5;
    const int waveId  = blockIdx.x * WAVES_PER_BLOCK + warp;
    const int numWaves = gridDim.x * WAVES_PER_BLOCK;
    const int numTiles = numPairs / TILE_PAIRS;

    // WMMA f32 16x16x4 A-layout roles: lane = (m, khalf)
    const int m  = lane & 15;      // pair (row of A) owned by this lane
    const int kh = lane >> 4;      // K-half: vgpr0/1 hold K = 2*kh, 2*kh+1

    float bias = 0.0f;
    if (mode) {
        float hb = h_bias[0];
        bias = log1pf(__expf(hb));          // softplus(h_bias)
    }

    float local = 0.0f;
    float* ws = &smem[warp][0][0];

    // B = all-ones 4x16 matrix: every lane's 2 VGPRs are 1.0
    v2f bones; bones.x = 1.0f; bones.y = 1.0f;

    for (int tile = waveId; tile < numTiles; tile += numWaves) {
        // ---- stage: each lane owns one of the tile's 32 rows -------------
        int row = pairIdx[tile * ROWS_PER_TILE + lane];
        if (tile + numWaves < numTiles)  // uniform branch; keeps EXEC full
            __builtin_prefetch(&pairIdx[(tile + numWaves) * ROWS_PER_TILE + lane], 0, 0);

        const float4* src = (const float4*)(Xemb + (size_t)row * D_DIM);
        float*        dst = ws + lane * LDS_STRIDE;
#pragma unroll 8
        for (int it = 0; it < 32; ++it) {           // 32 x b128 = 512 B row
            float4 v = src[it];
            *(float4*)(dst + it * 4) = v;
        }
        // Same-wave DS ops are in-order, but make the store->load edge explicit.
        asm volatile("s_wait_dscnt 0" ::: "memory");

        // ---- compute: 32 chained WMMAs over K = 0..127 -------------------
        const float* xi = ws + (2 * m)     * LDS_STRIDE;   // pair m, row i
        const float* xj = ws + (2 * m + 1) * LDS_STRIDE;   // pair m, row j

        v8f acc = {0.f, 0.f, 0.f, 0.f, 0.f, 0.f, 0.f, 0.f};
#pragma unroll 8
        for (int c = 0; c < 32; ++c) {
            int k0 = 4 * c + 2 * kh;                 // this lane's 2 K-slots
            float2 a2 = *(const float2*)(xi + k0);   // ds_load_b64
            float2 b2 = *(const float2*)(xj + k0);   // ds_load_b64
            float d0 = a2.x - b2.x;
            float d1 = a2.y - b2.y;
            v2f afrag; afrag.x = d0 * d0; afrag.y = d1 * d1;
            // D = A x ones + C  ->  acc[r] = partial dist^2, broadcast over N
            acc = __builtin_amdgcn_wmma_f32_16x16x4_f32(
                      /*neg_a=*/false, afrag, /*neg_b=*/false, bones,
                      /*c_mod=*/(short)0, acc, /*reuse_a=*/false, /*reuse_b=*/false);
        }

        // ---- per-pair post-op (values replicated across 16 lanes) --------
        if (mode == 0) {
            local += acc[0] + acc[1] + acc[2] + acc[3]
                   + acc[4] + acc[5] + acc[6] + acc[7];
        } else {
#pragma unroll
            for (int r = 0; r < 8; ++r) {
                float d = sqrtf(acc[r]);
                float t = bias - d;
                t = fmaxf(t, 0.0f);
                local += t * t;
            }
        }
    }

    // 0.5 * mean, and /16 because each pair's value is replicated on 16 lanes
    local *= 0.5f / ((float)numPairs * 16.0f);

    // wave32 reduction, one atomic per wave
    for (int off = 16; off > 0; off >>= 1)
        local += __shfl_down(local, off, 32);
    if (lane == 0)
        atomicAdd(&out[mode], local);
}

extern "C" void kernel_launch(void* const* d_in, const int* in_sizes, int n_in,
                              void* d_out, int out_size, void* d_ws, size_t ws_size,
                              hipStream_t stream) {
    const float* Xemb    = (const float*)d_in[0];
    // d_in[1] = scores (unused by forward), d_in[3] = labels (unused by forward)
    const float* h_bias  = (const float*)d_in[2];
    const int*   pos_idx = (const int*)d_in[4];
    const int*   neg_idx = (const int*)d_in[5];
    float* out = (float*)d_out;

    const int numPairs = in_sizes[4] / 2;   // 262144

    // zero the two accumulators (capture-safe)
    hipMemsetAsync(out, 0, 2 * sizeof(float), stream);

    dim3 grid(1024), block(BLOCK_THREADS);  // 2048 waves -> 8 tiles/wave/loss
    contrastive_wmma_kernel<<<grid, block, 0, stream>>>(Xemb, pos_idx, h_bias, out, numPairs, 0);
    contrastive_wmma_kernel<<<grid, block, 0, stream>>>(Xemb, neg_idx, h_bias, out, numPairs, 1);
}